// CustomGNN_5394478924556
// MI455X (gfx1250) — compile-verified
//
#include <hip/hip_runtime.h>

typedef __attribute__((ext_vector_type(2))) float v2f;
typedef __attribute__((ext_vector_type(8))) float v8f;

#define NNODES 50000
#define EPT    300000
#define NT32   ((NNODES + 31) / 32)   // 1563 node tiles (last has 16 valid rows)
#define ET32   (EPT / 32)             // 9375 edge tiles (exact)
#define NPAD   (NT32 * 32)            // 50016 padded node rows in workspace

// Intra-wave fence for cross-lane communication through LDS.
// LDS ops are DScnt-tracked (CDNA5 ISA §7.1); dscnt==0 makes all of this
// wave's LDS writes visible to subsequent LDS reads by any lane.
__device__ __forceinline__ void wave_lds_fence() {
  asm volatile("s_wait_dscnt 0x0" ::: "memory");
}

// B fragment (4x16, K x N) for V_WMMA_F32_16X16X4_F32.
// Lane l, vgpr v holds B[k0 + (l>>4)*2 + v][n0 + (l&15)].
// W is row-major [NACT][KACT] (out x in); B = W^T.
template<int NACT, int KACT>
__device__ __forceinline__ v2f load_b(const float* __restrict__ W, int n0, int k0, int lane) {
  int n = n0 + (lane & 15);
  int k = k0 + ((lane >> 4) << 1);
  v2f b = {0.f, 0.f};
  if (n < NACT) {
    if (k     < KACT) b.x = W[(size_t)n * KACT + k];
    if (k + 1 < KACT) b.y = W[(size_t)n * KACT + k + 1];
  }
  return b;
}

template<int NT>
__device__ __forceinline__ void acc_zero(v8f (&acc)[2][NT]) {
  const v8f vzero = {0.f, 0.f, 0.f, 0.f, 0.f, 0.f, 0.f, 0.f};
#pragma unroll
  for (int h = 0; h < 2; ++h)
#pragma unroll
    for (int t = 0; t < NT; ++t) acc[h][t] = vzero;
}

// Accumulate over a K range: two 16-row A fragments share every B fragment
// (2 WMMAs per weight fetch). aload(mbase, k0) yields the A fragment for rows
// [t0+mbase, t0+mbase+16) at depth k0.
template<int NT, int NACT, int KACT, class AF>
__device__ __forceinline__ void mma_accum(v8f (&acc)[2][NT], int kbeg, int kend,
    AF&& aload, const float* __restrict__ W, int lane) {
  for (int k0 = kbeg; k0 < kend; k0 += 4) {
    v2f a0 = aload(0,  k0);
    v2f a1 = aload(16, k0);
#pragma unroll
    for (int t = 0; t < NT; ++t) {
      v2f bf = load_b<NACT, KACT>(W, t * 16, k0, lane);
      acc[0][t] = __builtin_amdgcn_wmma_f32_16x16x4_f32(
          false, a0, false, bf, (short)0, acc[0][t], false, false);
      acc[1][t] = __builtin_amdgcn_wmma_f32_16x16x4_f32(
          false, a1, false, bf, (short)0, acc[1][t], false, false);
    }
  }
}

// Epilogue: bias (+ optional ReLU), store 32-row tile to LDS in C/D layout.
template<int NT, int NACT, bool RELU>
__device__ __forceinline__ void store_y(const v8f (&acc)[2][NT],
    const float* __restrict__ bias, float* __restrict__ Y, int lane) {
  int nl = lane & 15;
  int mo = (lane >> 4) << 3;   // D layout: vgpr r -> row (lane<16 ? r : r+8)
#pragma unroll
  for (int h = 0; h < 2; ++h)
#pragma unroll
    for (int t = 0; t < NT; ++t) {
      int n = t * 16 + nl;
      float bb = (n < NACT) ? bias[n] : 0.f;
#pragma unroll
      for (int r = 0; r < 8; ++r) {
        float v = acc[h][t][r] + bb;
        if (RELU) v = fmaxf(v, 0.f);
        if (n >= NACT) v = 0.f;
        Y[(h * 16 + mo + r) * 128 + n] = v;
      }
    }
  wave_lds_fence();
}

template<int KPAD, int NT, int NACT, int KACT, bool RELU, class AF>
__device__ __forceinline__ void run_layer32(AF&& aload,
    const float* __restrict__ W, const float* __restrict__ bias,
    float* __restrict__ Y, int lane) {
  v8f acc[2][NT];
  acc_zero<NT>(acc);
  mma_accum<NT, NACT, KACT>(acc, 0, KPAD, aload, W, lane);
  store_y<NT, NACT, RELU>(acc, bias, Y, lane);
}

// LayerNorm over 128 features: 32 rows onto 32 lanes (eps=1e-5, population var).
__device__ __forceinline__ void ln32(float* __restrict__ Y,
    const float* __restrict__ g, const float* __restrict__ beta, int lane) {
  float* row = Y + lane * 128;
  float mu = 0.f;
  for (int c = 0; c < 128; ++c) mu += row[c];
  mu *= (1.f / 128.f);
  float var = 0.f;
  for (int c = 0; c < 128; ++c) { float d = row[c] - mu; var += d * d; }
  var *= (1.f / 128.f);
  float inv = 1.f / sqrtf(var + 1e-5f);
  for (int c = 0; c < 128; ++c) row[c] = (row[c] - mu) * inv * g[c] + beta[c];
  wave_lds_fence();
}

// A fragment loader from a [32][128] LDS tile.
#define ALDS32(X) [&](int mbase, int k0) -> v2f {                       \
    int m_ = mbase + (lane & 15); int k_ = k0 + ((lane >> 4) << 1);     \
    v2f a_; a_.x = (X)[m_ * 128 + k_]; a_.y = (X)[m_ * 128 + k_ + 1];   \
    return a_; }

__global__ void k_zero(float* __restrict__ p, long n) {
  long i = (long)blockIdx.x * blockDim.x + threadIdx.x;
  long stride = (long)gridDim.x * blockDim.x;
  for (; i < n; i += stride) p[i] = 0.f;
}

// ---------- node encoder: [N,13] -> MLP(13,128,128,128)+LN -> nodes ----------
// Writes all NPAD rows (padding rows get deterministic values from zero-filled
// input reads) so downstream kernels never need row guards.
__global__ void __launch_bounds__(32)
k_encode_nodes(const float* __restrict__ nf,
               const float* __restrict__ W0, const float* __restrict__ W1, const float* __restrict__ W2,
               const float* __restrict__ b0, const float* __restrict__ b1, const float* __restrict__ b2,
               const float* __restrict__ g,  const float* __restrict__ beta,
               float* __restrict__ nodes) {
  __shared__ float A[32 * 128];
  __shared__ float B[32 * 128];
  const int lane = threadIdx.x & 31;
  const int t0 = blockIdx.x * 32;

  run_layer32<16, 8, 128, 13, true>([&](int mbase, int k0) -> v2f {
      int m = mbase + (lane & 15);
      int r = t0 + m;
      v2f a = {0.f, 0.f};
      if (r < NNODES) {                       // only guard on raw input reads
        int k = k0 + ((lane >> 4) << 1);
        const float* row = nf + (size_t)r * 13;
        if (k < 13)     a.x = row[k];
        if (k + 1 < 13) a.y = row[k + 1];
      }
      return a;
    }, W0, b0, A, lane);
  run_layer32<128, 8, 128, 128, true >(ALDS32(A), W1, b1, B, lane);
  run_layer32<128, 8, 128, 128, false>(ALDS32(B), W2, b2, A, lane);
  ln32(A, g, beta, lane);

  for (int m = 0; m < 32; ++m) {             // unguarded: nodes has NPAD rows
    float4 v = *(const float4*)&A[m * 128 + lane * 4];
    *(float4*)&nodes[(size_t)(t0 + m) * 128 + (size_t)lane * 4] = v;
  }
}

// ---- edge encoder fused with scatter: [E,4] -> MLP+LN -> atomicAdd agg -----
__global__ void __launch_bounds__(32)
k_encode_edges(const float* __restrict__ ef,
               const float* __restrict__ W0, const float* __restrict__ W1, const float* __restrict__ W2,
               const float* __restrict__ b0, const float* __restrict__ b1, const float* __restrict__ b2,
               const float* __restrict__ g,  const float* __restrict__ beta,
               const int* __restrict__ recv, int recv_off, int etype,
               float* __restrict__ agg) {
  __shared__ float A[32 * 128];
  __shared__ float B[32 * 128];
  const int lane = threadIdx.x & 31;
  const int t0 = blockIdx.x * 32;   // EPT % 32 == 0: no guards needed

  run_layer32<4, 8, 128, 4, true>([&](int mbase, int k0) -> v2f {
      int m = mbase + (lane & 15);
      int k = k0 + ((lane >> 4) << 1);
      const float* row = ef + (size_t)(t0 + m) * 4;
      v2f a; a.x = row[k]; a.y = row[k + 1]; return a;
    }, W0, b0, A, lane);
  run_layer32<128, 8, 128, 128, true >(ALDS32(A), W1, b1, B, lane);
  run_layer32<128, 8, 128, 128, false>(ALDS32(B), W2, b2, A, lane);
  ln32(A, g, beta, lane);

  const int c0 = lane * 4;
  for (int m = 0; m < 32; ++m) {
    int seg = recv[recv_off + t0 + m] * 3 + etype;
    float* dst = agg + (size_t)seg * 128 + c0;
    const float* src = &A[m * 128 + c0];
    atomicAdd(dst + 0, src[0]);
    atomicAdd(dst + 1, src[1]);
    atomicAdd(dst + 2, src[2]);
    atomicAdd(dst + 3, src[3]);
  }
}

// -------- MP step: nodes = MLP_upd(concat(nodes, agg.reshape(N,384))) --------
// Fully unguarded (padded ws buffers); concat handled as two K ranges over the
// same accumulators, so the hot loop is pure clause-loads + WMMA.
__global__ void __launch_bounds__(32)
k_mp_step(const float* __restrict__ nodes_in, const float* __restrict__ agg,
          const float* __restrict__ W0, const float* __restrict__ W1, const float* __restrict__ W2,
          const float* __restrict__ b0, const float* __restrict__ b1, const float* __restrict__ b2,
          const float* __restrict__ g,  const float* __restrict__ beta,
          float* __restrict__ nodes_out) {
  __shared__ float A[32 * 128];
  __shared__ float B[32 * 128];
  const int lane = threadIdx.x & 31;
  const int t0 = blockIdx.x * 32;

  v8f acc[2][8];
  acc_zero<8>(acc);
  mma_accum<8, 128, 512>(acc, 0, 128, [&](int mbase, int k0) -> v2f {
      int m = mbase + (lane & 15);
      int k = k0 + ((lane >> 4) << 1);
      const float* p = nodes_in + (size_t)(t0 + m) * 128 + k;
      v2f a; a.x = p[0]; a.y = p[1]; return a;
    }, W0, lane);
  mma_accum<8, 128, 512>(acc, 128, 512, [&](int mbase, int k0) -> v2f {
      int m = mbase + (lane & 15);
      int k = k0 + ((lane >> 4) << 1);
      const float* p = agg + (size_t)(t0 + m) * 384 + (k - 128);
      v2f a; a.x = p[0]; a.y = p[1]; return a;
    }, W0, lane);
  store_y<8, 128, true>(acc, b0, A, lane);

  run_layer32<128, 8, 128, 128, true >(ALDS32(A), W1, b1, B, lane);
  run_layer32<128, 8, 128, 128, false>(ALDS32(B), W2, b2, A, lane);
  ln32(A, g, beta, lane);

  for (int m = 0; m < 32; ++m) {
    float4 v = *(const float4*)&A[m * 128 + lane * 4];
    *(float4*)&nodes_out[(size_t)(t0 + m) * 128 + (size_t)lane * 4] = v;
  }
}

// ------------------- decoder: nodes -> MLP(128,128,128,3) -------------------
__global__ void __launch_bounds__(32)
k_decode(const float* __restrict__ nodes,
         const float* __restrict__ W0, const float* __restrict__ W1, const float* __restrict__ W2,
         const float* __restrict__ b0, const float* __restrict__ b1, const float* __restrict__ b2,
         float* __restrict__ out) {
  __shared__ float A[32 * 128];
  __shared__ float B[32 * 128];
  const int lane = threadIdx.x & 31;
  const int t0 = blockIdx.x * 32;

  run_layer32<128, 8, 128, 128, true>([&](int mbase, int k0) -> v2f {
      int m = mbase + (lane & 15);
      int k = k0 + ((lane >> 4) << 1);
      const float* p = nodes + (size_t)(t0 + m) * 128 + k;   // padded ws: no guard
      v2f a; a.x = p[0]; a.y = p[1]; return a;
    }, W0, b0, A, lane);
  run_layer32<128, 8, 128, 128, true>(ALDS32(A), W1, b1, B, lane);
  run_layer32<128, 1,   3, 128, false>(ALDS32(B), W2, b2, A, lane);

  int r = t0 + lane;                 // 32 rows onto 32 lanes; d_out is exact-size
  if (r < NNODES) {
    out[(size_t)r * 3 + 0] = A[lane * 128 + 0];
    out[(size_t)r * 3 + 1] = A[lane * 128 + 1];
    out[(size_t)r * 3 + 2] = A[lane * 128 + 2];
  }
}

extern "C" void kernel_launch(void* const* d_in, const int* in_sizes, int n_in,
                              void* d_out, int out_size, void* d_ws, size_t ws_size,
                              hipStream_t stream) {
  const float* const* P = (const float* const*)d_in;
  const float* node_feats  = P[0];
  const float* body_feats  = P[1];
  const float* cable_feats = P[2];
  const float* con_feats   = P[3];
  // d_in[4] = senders: unused by the reference math (seg_ids uses receivers only)
  const int* receivers = (const int*)d_in[5];
  // Per-MLP dict flattened in insertion order: W0,W1,W2,b0,b1,b2[,g,beta]
  // node_enc@6, body_enc@14, cable_enc@22, con_enc@30, node_upd@38, dec@46 (no g/beta)
#define MLP8(base) P[base], P[base+1], P[base+2], P[base+3], P[base+4], P[base+5], P[base+6], P[base+7]

  float* agg   = (float*)d_ws;                     // [3*NPAD][128]
  float* nodes = agg + (size_t)3 * NPAD * 128;     // [NPAD][128]

  k_zero<<<2048, 256, 0, stream>>>(agg, (long)3 * NPAD * 128);

  const dim3 blk(32);   // one wave32 per block; intra-wave LDS sync only

  k_encode_nodes<<<NT32, blk, 0, stream>>>(node_feats, MLP8(6), nodes);

  // agg is loop-invariant in the reference: compute encode + segment-sum once.
  k_encode_edges<<<ET32, blk, 0, stream>>>(body_feats,  MLP8(14), receivers, 0,       0, agg);
  k_encode_edges<<<ET32, blk, 0, stream>>>(cable_feats, MLP8(22), receivers, EPT,     1, agg);
  k_encode_edges<<<ET32, blk, 0, stream>>>(con_feats,   MLP8(30), receivers, 2 * EPT, 2, agg);

  for (int s = 0; s < 10; ++s)
    k_mp_step<<<NT32, blk, 0, stream>>>(nodes, agg, MLP8(38), nodes);  // row-local => in-place safe

  k_decode<<<NT32, blk, 0, stream>>>(nodes, P[46], P[47], P[48], P[49], P[50], P[51],
                                     (float*)d_out);
#undef MLP8
}